// DenseWarp_13374528160262
// MI455X (gfx1250) — compile-verified
//
#include <hip/hip_runtime.h>

// dense_image_warp (bilinear backward warp), B=4 C=4 H=1080 W=1920, fp32.
// Memory-bound gather: ~332 MB minimum HBM traffic -> ~14 us at 23.3 TB/s.
// CDNA5 paths: global_prefetch_b8 (next-row flow read-ahead), TH_NT loads
// (flow, read-once) and TH_NT b128 stores (out, write-once) so the 192 MB L2
// stays resident with the 132.7 MB frame for the data-dependent gathers.
//
// Geometry (no integer division anywhere):
//   block = 256 threads = 8 wave32 waves
//   lane (0..31) covers 4 consecutive x  -> wave spans 128 x
//   wave (0..7)  covers 3 consecutive y  -> block tile = 128 x * 24 y
//   grid = (1920/128, 1080/24, B) = (15, 45, 4)   [all divide exactly]
//
// Edge handling is folded into the weights (no per-channel selects):
//   xa = min(x0, W-2); pair = {row[xa], row[xa+1]}.
//   x0 == W-1 implies qx == W-1 exactly -> correct sample is pair.y,
//   achieved by forcing wx = 1. y-clamp needs nothing: y1 clamps only when
//   wy == 0, where re-loading row0 as row1 is already correct.

#define HH 1080
#define WW 1920
#define CC 4

static constexpr int HW    = HH * WW;   // 2,073,600 (channel-plane stride)
static constexpr int PX    = 4;         // pixels per thread (x)
static constexpr int ROWS  = 3;         // rows per wave
static constexpr int TILEW = 32 * PX;   // 128
static constexpr int TILEH = 8 * ROWS;  // 24
static constexpr int OLAST = (HH - 1) * WW;  // offset of last row start

// clang native vector (required by __builtin_nontemporal_load/store)
typedef float v4f __attribute__((ext_vector_type(4)));

__global__ __launch_bounds__(256)
void DenseWarp_kernel(const float* __restrict__ frame,
                      const float* __restrict__ flow,
                      float* __restrict__ out) {
    const int lane = threadIdx.x & 31;
    const int wave = threadIdx.x >> 5;
    const int x0px = blockIdx.x * TILEW + lane * PX;       // first of 4 x
    const int ybeg = blockIdx.y * TILEH + wave * ROWS;     // first of 3 y
    const int b    = blockIdx.z;

    const float* fbase  = frame + (size_t)b * CC * HW;     // frame[b]
    const float* flbase = flow  + (size_t)b * 2  * HW;     // flow[b]
    float*       obase  = out   + (size_t)b * CC * HW;     // out[b]

    #pragma unroll
    for (int r = 0; r < ROWS; ++r) {
        const int y = ybeg + r;
        const float* fl = flbase + (size_t)y * WW + x0px;

        // read-ahead for the next row of this wave (gfx1250 global_prefetch_b8)
        if (r + 1 < ROWS) {
            __builtin_prefetch(fl + WW, 0, 1);        // flow_y, next row
            __builtin_prefetch(fl + WW + HW, 0, 1);   // flow_x, next row
        }

        // flow fetch: read-once streams, non-temporal b128 loads (16B aligned)
        const v4f fy4 = __builtin_nontemporal_load((const v4f*)fl);
        const v4f fx4 = __builtin_nontemporal_load((const v4f*)(fl + HW));

        // per-pixel sampling setup, reused across all 4 channels
        float wxv[PX], wyv[PX], omwx[PX], omwy[PX];
        int   o0[PX], o1[PX];
        #pragma unroll
        for (int k = 0; k < PX; ++k) {
            const float qy = fminf(fmaxf((float)y - fy4[k], 0.0f), (float)(HH - 1));
            const float qx = fminf(fmaxf((float)(x0px + k) - fx4[k], 0.0f), (float)(WW - 1));
            const float y0f = floorf(qy);
            const float x0f = floorf(qx);
            const int iy0 = (int)y0f;
            const int ix0 = (int)x0f;
            const int xa  = min(ix0, WW - 2);
            // weight-folded right-edge clamp: x0==W-1 -> wx:=1 selects pair.y
            const float wx = (ix0 > xa) ? 1.0f : (qx - x0f);
            wxv[k]  = wx;
            omwx[k] = 1.0f - wx;
            wyv[k]  = qy - y0f;
            omwy[k] = 1.0f - wyv[k];
            o0[k] = iy0 * WW + xa;
            o1[k] = min(o0[k] + WW, xa + OLAST);   // y-clamped second row
        }

        #pragma unroll
        for (int c = 0; c < CC; ++c) {
            const float* pc = fbase + (size_t)c * HW;
            v4f rr;
            #pragma unroll
            for (int k = 0; k < PX; ++k) {
                float2 p0, p1;                  // one b64 gather per corner row
                __builtin_memcpy(&p0, pc + o0[k], sizeof(float2));
                __builtin_memcpy(&p1, pc + o1[k], sizeof(float2));
                const float top = fmaf(wxv[k], p0.y, omwx[k] * p0.x);
                const float bot = fmaf(wxv[k], p1.y, omwx[k] * p1.x);
                rr[k] = fmaf(wyv[k], bot, omwy[k] * top);
            }
            // write-once output stream: one non-temporal b128 store per channel
            float* o = obase + (size_t)c * HW + (size_t)y * WW + x0px;
            __builtin_nontemporal_store(rr, (v4f*)o);
        }
    }
}

extern "C" void kernel_launch(void* const* d_in, const int* in_sizes, int n_in,
                              void* d_out, int out_size, void* d_ws, size_t ws_size,
                              hipStream_t stream) {
    const float* frame = (const float*)d_in[0];  // [B,C,H,W] fp32
    const float* flow  = (const float*)d_in[1];  // [B,2,H,W] fp32
    float* out = (float*)d_out;                  // [B,C,H,W] fp32

    const dim3 block(256);                       // 8 wave32 waves
    const dim3 grid(WW / TILEW, HH / TILEH, 4);  // (15, 45, 4) — exact tiling
    DenseWarp_kernel<<<grid, block, 0, stream>>>(frame, flow, out);
}